// Attention_22162031247906
// MI455X (gfx1250) — compile-verified
//
#include <hip/hip_runtime.h>
#include <hip/hip_bf16.h>

// Problem sizes (match the reference)
#define HIDDEN 1024
#define EMBED  512
#define BATCH  32
#define SEQ    2048
#define NCHUNK 16                 // L-chunks; 128 rows (512 KB of keys) per chunk
#define CROWS  (SEQ / NCHUNK)     // 128

typedef __attribute__((ext_vector_type(2))) float v2f;
typedef __attribute__((ext_vector_type(8))) float v8f;
typedef __attribute__((ext_vector_type(4))) unsigned int v4u;
typedef __attribute__((ext_vector_type(4))) int v4i;
typedef __attribute__((ext_vector_type(8))) int v8i;

// ---------------------------------------------------------------------------
// TDM: DMA one contiguous row of `nelem` fp32 from global into LDS offset
// `lds_off`.  D# built per CDNA5 ISA §8: group0 {count=1, lds_addr,
// global_addr, type=2}, group1 {data_size=4B, tensor_dim0=nelem,
// tensor_dim1=1, tile_dim0=nelem, tile_dim1=1, dim0_stride=nelem}.
// Issue from ONE wave; follow with s_wait_tensorcnt + workgroup barrier.
// ---------------------------------------------------------------------------
__device__ __forceinline__ void tdm_load_row_f32(const float* gsrc,
                                                 unsigned lds_off, int nelem) {
    unsigned long long ga = (unsigned long long)(uintptr_t)gsrc;
    v4u g0 = { 1u,                                   // count=1 (valid D#)
               lds_off,                               // lds_addr (bytes)
               (unsigned)(ga & 0xFFFFFFFFu),          // global_addr[31:0]
               (unsigned)((ga >> 32) & 0x1FFFFFFu) | (2u << 30) }; // [56:32]|type=2
    v8i g1 = { (int)(2u << 16),                       // data_size=2 -> 4 bytes
               (int)((unsigned)(nelem & 0xFFFF) << 16),          // tensor_dim0 lo16
               (int)(((unsigned)nelem >> 16) | (1u << 16)),      // dim0 hi16 | tensor_dim1=1
               (int)((unsigned)(nelem & 0xFFFF) << 16),          // tile_dim0
               1,                                                // tile_dim1=1
               nelem,                                            // dim0_stride lo32
               0, 0 };
    v4i z4 = {0, 0, 0, 0};
#if __clang_major__ >= 23
    v8i z8 = {0, 0, 0, 0, 0, 0, 0, 0};
    __builtin_amdgcn_tensor_load_to_lds(g0, g1, z4, z4, z8, 0);
#else
    __builtin_amdgcn_tensor_load_to_lds(g0, g1, z4, z4, 0);
#endif
}

// ---------------------------------------------------------------------------
// Stage 1: q[b,o] = sum_h query[b,h] * Wq[o,h]   (M=32, N=1024, K=1024)
// One wave per 16x16 tile, V_WMMA_F32_16X16X4_F32 (exact fp32).
// ---------------------------------------------------------------------------
__global__ void qproj_wmma(const float* __restrict__ query,
                           const float* __restrict__ Wq,
                           float* __restrict__ q) {
    const int tileM = (blockIdx.x & 1) * 16;
    const int tileN = (blockIdx.x >> 1) * 16;
    const int lane  = threadIdx.x;
    const int lm    = lane & 15;
    const int kh    = (lane >> 4) << 1;               // 0 or 2

    const float* arow = query + (size_t)(tileM + lm) * HIDDEN;
    const float* brow = Wq    + (size_t)(tileN + lm) * HIDDEN;

    v8f c = {};
    for (int k = 0; k < HIDDEN; k += 4) {
        v2f a = *(const v2f*)(arow + k + kh);
        v2f b = *(const v2f*)(brow + k + kh);
        c = __builtin_amdgcn_wmma_f32_16x16x4_f32(false, a, false, b,
                                                  (short)0, c, false, false);
    }
    const int mbase = (lane >> 4) * 8;
    for (int r = 0; r < 8; ++r)
        q[(size_t)(tileM + mbase + r) * HIDDEN + tileN + lm] = c[r];
}

// ---------------------------------------------------------------------------
// Stage 2 (fused, flash-style): per (b, chunk of 128 rows):
//   phase 0: TDM q[b,:] -> LDS
//   phase 1: scores s_l = keys[b,l,:].q   (keys chunk: HBM read #1 and only)
//   local softmax pieces: m_c = max s_l ; e_l = exp(s_l-m_c) ; S_c = sum e_l
//   phase 2: pk[b,c,h] = sum_l e_l * keys[b,l,h]   (chunk re-read, L2-hot)
//   phase 3: pe[b,c,e] = sum_l e_l * enc[b,l,e]
// Combine kernel later rescales by exp(m_c - M)/D.  Deterministic.
// ---------------------------------------------------------------------------
struct FusedSmem {
    float4 qs4[HIDDEN / 4];   // MUST stay first: TDM writes LDS offset 0
    float  ssh[CROWS];
    float  esh[CROWS];
};

__global__ void fused_scores_wsum(const float* __restrict__ keys,
                                  const float* __restrict__ enc,
                                  const float* __restrict__ q,
                                  float2* __restrict__ mS,
                                  float* __restrict__ pk,
                                  float* __restrict__ pe) {
    __shared__ FusedSmem sm;
    const int b     = blockIdx.x / NCHUNK;
    const int chunk = blockIdx.x % NCHUNK;
    const int lbase = chunk * CROWS;
    const int t     = threadIdx.x;
    const int wave  = t >> 5;
    const int lane  = t & 31;

    // phase 0: q row -> LDS via Tensor Data Mover (wave 0 issues & waits)
    if (t < 32) {
        tdm_load_row_f32(q + (size_t)b * HIDDEN, 0u, HIDDEN);
        __builtin_amdgcn_s_wait_tensorcnt(0);
    }
    __syncthreads();

    // phase 1: 8 waves x 16 rows; lane-parallel dot + shuffle reduction
    const float4* kbase4 =
        (const float4*)(keys + ((size_t)b * SEQ + lbase) * HIDDEN);
    for (int r = 0; r < CROWS / 8; ++r) {
        const int row = wave * (CROWS / 8) + r;
        const float4* krow = kbase4 + (size_t)row * (HIDDEN / 4);
        float acc = 0.f;
        #pragma unroll
        for (int it = 0; it < HIDDEN / 128; ++it) {
            const int idx = lane + it * 32;
            float4 kv = krow[idx];
            float4 qv = sm.qs4[idx];
            acc += kv.x * qv.x + kv.y * qv.y + kv.z * qv.z + kv.w * qv.w;
        }
        for (int off = 16; off > 0; off >>= 1) acc += __shfl_xor(acc, off, 32);
        if (lane == 0) sm.ssh[row] = acc;
    }
    __syncthreads();

    // local max (redundant broadcast scan: cheap, barrier-free)
    float m = -INFINITY;
    for (int i = 0; i < CROWS; ++i) m = fmaxf(m, sm.ssh[i]);
    if (t < CROWS) sm.esh[t] = expf(sm.ssh[t] - m);
    __syncthreads();
    if (t == 0) {
        float S = 0.f;
        for (int i = 0; i < CROWS; ++i) S += sm.esh[i];
        mS[(size_t)b * NCHUNK + chunk] = make_float2(m, S);
    }

    // phase 2: weighted keys columns (4 per thread; chunk is L2-resident)
    {
        float4 acc = {0.f, 0.f, 0.f, 0.f};
        for (int i = 0; i < CROWS; ++i) {
            if (i + 4 < CROWS)
                __builtin_prefetch(&kbase4[(size_t)(i + 4) * 256 + t], 0, 0);
            const float  w = sm.esh[i];
            const float4 v = kbase4[(size_t)i * 256 + t];
            acc.x += w * v.x; acc.y += w * v.y; acc.z += w * v.z; acc.w += w * v.w;
        }
        ((float4*)(pk + ((size_t)b * NCHUNK + chunk) * HIDDEN))[t] = acc;
    }

    // phase 3: weighted enc columns (2 per thread; enc read once from HBM)
    {
        const float2* ebase2 =
            (const float2*)(enc + ((size_t)b * SEQ + lbase) * EMBED);
        float2 acc = {0.f, 0.f};
        for (int i = 0; i < CROWS; ++i) {
            if (i + 4 < CROWS)
                __builtin_prefetch(&ebase2[(size_t)(i + 4) * 256 + t], 0, 0);
            const float  w = sm.esh[i];
            const float2 v = ebase2[(size_t)i * 256 + t];
            acc.x += w * v.x; acc.y += w * v.y;
        }
        ((float2*)(pe + ((size_t)b * NCHUNK + chunk) * EMBED))[t] = acc;
    }
}

// ---------------------------------------------------------------------------
// Stage 3: combine chunk partials with softmax rescale:
//   M = max_c m_c ; D = sum_c exp(m_c-M)*S_c
//   dst[b,c] = sum_c exp(m_c-M)*part[b,c,:] / D
// ---------------------------------------------------------------------------
__global__ void combine_partials(const float* __restrict__ part,
                                 const float2* __restrict__ mS,
                                 float* __restrict__ dst, int width) {
    const int idx = blockIdx.x * blockDim.x + threadIdx.x;  // 0..B*width-1
    const int b = idx / width, c = idx % width;

    float M = -INFINITY;
    #pragma unroll
    for (int k = 0; k < NCHUNK; ++k)
        M = fmaxf(M, mS[(size_t)b * NCHUNK + k].x);
    float D = 0.f, s = 0.f;
    #pragma unroll
    for (int k = 0; k < NCHUNK; ++k) {
        const float2 ms = mS[(size_t)b * NCHUNK + k];
        const float  r  = expf(ms.x - M);
        D += r * ms.y;
        s += r * part[((size_t)b * NCHUNK + k) * width + c];
    }
    dst[idx] = s / D;
}

// ---------------------------------------------------------------------------
// Stage 4: out[b,o] = sum_e wenc[b,e]*Wcs[e,o] + sum_h wkeys[b,h]*Wkey[o,h]
// ---------------------------------------------------------------------------
__global__ void outproj_wmma(const float* __restrict__ wenc,
                             const float* __restrict__ wkeys,
                             const float* __restrict__ Wcs,
                             const float* __restrict__ Wkey,
                             float* __restrict__ out) {
    const int tileM = (blockIdx.x & 1) * 16;
    const int tileN = (blockIdx.x >> 1) * 16;
    const int lane  = threadIdx.x;
    const int lm    = lane & 15;
    const int kh    = (lane >> 4) << 1;

    v8f c = {};

    const float* aE  = wenc + (size_t)(tileM + lm) * EMBED;
    const int    col = tileN + lm;
    for (int k = 0; k < EMBED; k += 4) {
        v2f a = *(const v2f*)(aE + k + kh);
        v2f b;
        b.x = Wcs[(size_t)(k + kh) * HIDDEN + col];
        b.y = Wcs[(size_t)(k + kh + 1) * HIDDEN + col];
        c = __builtin_amdgcn_wmma_f32_16x16x4_f32(false, a, false, b,
                                                  (short)0, c, false, false);
    }

    const float* aK = wkeys + (size_t)(tileM + lm) * HIDDEN;
    const float* bK = Wkey  + (size_t)(tileN + lm) * HIDDEN;
    for (int k = 0; k < HIDDEN; k += 4) {
        v2f a = *(const v2f*)(aK + k + kh);
        v2f b = *(const v2f*)(bK + k + kh);
        c = __builtin_amdgcn_wmma_f32_16x16x4_f32(false, a, false, b,
                                                  (short)0, c, false, false);
    }

    const int mbase = (lane >> 4) * 8;
    for (int r = 0; r < 8; ++r)
        out[(size_t)(tileM + mbase + r) * HIDDEN + tileN + lm] = c[r];
}

// ---------------------------------------------------------------------------
extern "C" void kernel_launch(void* const* d_in, const int* in_sizes, int n_in,
                              void* d_out, int out_size, void* d_ws, size_t ws_size,
                              hipStream_t stream) {
    const float* query = (const float*)d_in[0];   // [32,1,1024]
    const float* keys  = (const float*)d_in[1];   // [32,2048,1024]
    const float* enc   = (const float*)d_in[2];   // [32,2048,512]
    const float* Wq    = (const float*)d_in[3];   // [1024,1024]
    const float* Wk    = (const float*)d_in[4];   // [1024,1024]
    const float* Wcs   = (const float*)d_in[5];   // [512,1024]
    float* out = (float*)d_out;                   // [32,1024]

    float*  ws    = (float*)d_ws;                 // ~3.5 MB total
    float*  q     = ws;                                       // B*H
    float2* mS    = (float2*)(q + (size_t)BATCH * HIDDEN);    // B*NCHUNK pairs
    float*  pk    = (float*)(mS + (size_t)BATCH * NCHUNK);    // B*NCHUNK*H
    float*  pe    = pk + (size_t)BATCH * NCHUNK * HIDDEN;     // B*NCHUNK*E
    float*  wkeys = pe + (size_t)BATCH * NCHUNK * EMBED;      // B*H
    float*  wenc  = wkeys + (size_t)BATCH * HIDDEN;           // B*E

    // 1) q = query @ Wq^T                     (WMMA fp32)
    qproj_wmma<<<(BATCH / 16) * (HIDDEN / 16), 32, 0, stream>>>(query, Wq, q);
    // 2) fused scores + local softmax + weighted partial sums
    //    (keys: one HBM pass; enc: one HBM pass; TDM stages q into LDS)
    fused_scores_wsum<<<BATCH * NCHUNK, 256, 0, stream>>>(keys, enc, q,
                                                          mS, pk, pe);
    // 3) rescale + reduce chunk partials
    combine_partials<<<BATCH * HIDDEN / 256, 256, 0, stream>>>(pk, mS, wkeys, HIDDEN);
    combine_partials<<<BATCH * EMBED  / 256, 256, 0, stream>>>(pe, mS, wenc,  EMBED);
    // 4) out = wenc @ Wcs + wkeys @ Wk^T      (WMMA fp32)
    outproj_wmma<<<(BATCH / 16) * (HIDDEN / 16), 32, 0, stream>>>(wenc, wkeys,
                                                                  Wcs, Wk, out);
}